// CRFModel_13761075216767
// MI455X (gfx1250) — compile-verified
//
#include <hip/hip_runtime.h>
#include <hip/hip_bf16.h>

// CRF neg-mean-log-likelihood for B=256, L=2048, T=17.
//
// Denominator: scaled forward algorithm in LINEAR space.
//   a_t = diag(exp(em_t)) * W^T * a_{t-1},  W = exp(trans)  (17x17)
// Associative chain of 17x17 f32 matrix products -> chunked parallel scan.
// Each (batch, chunk) wave computes its chunk's 17x17 transfer matrix with
// V_WMMA_F32_16X16X4_F32 (M/N padded to 16+1 tiles, K padded to 20 with zero
// A columns; zero pads provably stay zero). Rescale every 8 steps keeps f32
// in range; log-scales accumulate on the side. Chunk count chosen from
// ws_size (32 chunks -> 8192 waves when workspace allows).

#define TNUM   17
#define LSEQ   2048

typedef float v2f __attribute__((ext_vector_type(2)));
typedef float v8f __attribute__((ext_vector_type(8)));

static __device__ __forceinline__ v8f vzero8() {
  v8f z = {0.f, 0.f, 0.f, 0.f, 0.f, 0.f, 0.f, 0.f};
  return z;
}

// D(16x16,f32) += A(16x4,f32) x B(4x16,f32)
static __device__ __forceinline__ v8f wmma4(v2f a, v2f b, v8f c) {
  return __builtin_amdgcn_wmma_f32_16x16x4_f32(false, a, false, b, (short)0, c,
                                               false, false);
}

// Build B operand (K across {vgpr, lane-half}, N across lanes) from P held in
// D layout (rows r lo-half / r+8 hi-half across vgprs, cols across lanes).
// S[] are half-swapped copies of P[0..7] (shfl_xor 16).
#define BUILD_B(B, P, S)                                                       \
  do {                                                                         \
    B[0].x = h ? S[2] : P[0]; B[0].y = h ? S[3] : P[1];                        \
    B[1].x = h ? S[6] : P[4]; B[1].y = h ? S[7] : P[5];                        \
    B[2].x = h ? P[2] : S[0]; B[2].y = h ? P[3] : S[1];                        \
    B[3].x = h ? P[6] : S[4]; B[3].y = h ? P[7] : S[5];                        \
    B[4].x = h ? 0.f : P[8];  B[4].y = 0.f;                                    \
  } while (0)

// ---------------------------------------------------------------------------
// Chunk kernel: one wave per (batch, chunk). Computes the 17x17 transfer
// matrix P = prod_{t in chunk} diag(e_t) * W^T (plus log-scale) -> 290 floats.
// ---------------------------------------------------------------------------
__global__ __launch_bounds__(32) void crf_chunk(
    const float* __restrict__ em, const unsigned char* __restrict__ mask,
    const float* __restrict__ trans, float* __restrict__ Pws, int clen) {
  const int b = blockIdx.x, c = blockIdx.y;
  const int nchunk = gridDim.y;
  const int lane = (int)threadIdx.x;
  const int h = lane >> 4;   // lane half
  const int ln = lane & 15;

  // A operand: Wt[j][k] = exp(trans[k][j]); rows j (M) across lanes,
  // K across (vgpr component, lane half). Pads are exactly 0.
  v2f A0[5], A1[5];
#pragma unroll
  for (int s = 0; s < 5; ++s) {
    const int k0 = 4 * s + 2 * h, k1 = k0 + 1;
    const int j0 = ln;       // M-tile 0: rows 0..15
    const int j1 = 16 + ln;  // M-tile 1: only row 16 valid
    const int ck0 = (k0 < TNUM) ? k0 : 0;
    const int ck1 = (k1 < TNUM) ? k1 : 0;
    const int cj1 = (j1 < TNUM) ? j1 : 0;
    A0[s].x = (k0 < TNUM) ? __expf(trans[ck0 * TNUM + j0]) : 0.f;
    A0[s].y = (k1 < TNUM) ? __expf(trans[ck1 * TNUM + j0]) : 0.f;
    A1[s].x = (k0 < TNUM && j1 < TNUM) ? __expf(trans[ck0 * TNUM + cj1]) : 0.f;
    A1[s].y = (k1 < TNUM && j1 < TNUM) ? __expf(trans[ck1 * TNUM + cj1]) : 0.f;
  }

  // P state in D layout. p0: N columns 0..15; p1: column 16 (lane 0 of tile).
  // p*[0..7]: rows r (lo half) / r+8 (hi half); p*[8]: row 16 (lo half).
  float p0[9], p1[9];
#pragma unroll
  for (int r = 0; r < 8; ++r) {
    p0[r] = ((h == 0 && ln == r) || (h == 1 && ln == r + 8)) ? 1.f : 0.f;
    p1[r] = 0.f;
  }
  p0[8] = 0.f;
  p1[8] = (lane == 0) ? 1.f : 0.f;  // identity (16,16)

  float ls = 0.f;
  const int t0 = 1 + c * clen;
  const int t1 = ((t0 + clen) < LSEQ) ? (t0 + clen) : LSEQ;
  const unsigned char* mk = mask + (size_t)b * LSEQ;

  for (int t = t0; t < t1; ++t) {
    if (mk[t]) {  // wave-uniform; masked steps are identity
      const float* emt = em + ((size_t)b * LSEQ + t) * TNUM;
      if (t + 1 < t1) __builtin_prefetch(emt + TNUM, 0, 1);
      float ev[8];
#pragma unroll
      for (int r = 0; r < 8; ++r) ev[r] = __expf(emt[r + 8 * h]);
      const float e16 = __expf(emt[16]);

      float s0[8], s1[8];
#pragma unroll
      for (int r = 0; r < 8; ++r) {
        s0[r] = __shfl_xor(p0[r], 16, 32);
        s1[r] = __shfl_xor(p1[r], 16, 32);
      }
      v2f B0[5], B1[5];
      BUILD_B(B0, p0, s0);
      BUILD_B(B1, p1, s1);

      v8f D00 = vzero8(), D01 = vzero8(), D10 = vzero8(), D11 = vzero8();
#pragma unroll
      for (int s = 0; s < 5; ++s) {
        D00 = wmma4(A0[s], B0[s], D00);  // rows 0..15, cols 0..15
        D01 = wmma4(A0[s], B1[s], D01);  // rows 0..15, col 16
        D10 = wmma4(A1[s], B0[s], D10);  // row 16,    cols 0..15
        D11 = wmma4(A1[s], B1[s], D11);  // row 16,    col 16
      }
      // scale rows by e (diag) and write back as next P
#pragma unroll
      for (int r = 0; r < 8; ++r) {
        p0[r] = D00[r] * ev[r];
        p1[r] = D01[r] * ev[r];
      }
      p0[8] = D10[0] * e16;
      p1[8] = D11[0] * e16;
    }
    if (((t - t0) & 7) == 7) {  // rescale every 8 steps
      float mx = p0[0];
#pragma unroll
      for (int r = 1; r < 9; ++r) mx = fmaxf(mx, p0[r]);
#pragma unroll
      for (int r = 0; r < 9; ++r) mx = fmaxf(mx, p1[r]);
#pragma unroll
      for (int off = 16; off > 0; off >>= 1)
        mx = fmaxf(mx, __shfl_xor(mx, off, 32));
      const float inv = 1.f / mx;
      ls += __logf(mx);
#pragma unroll
      for (int r = 0; r < 9; ++r) {
        p0[r] *= inv;
        p1[r] *= inv;
      }
    }
  }

  float* Pm = Pws + (size_t)(b * nchunk + c) * 290;
#pragma unroll
  for (int r = 0; r < 8; ++r) {
    Pm[(r + 8 * h) * TNUM + ln] = p0[r];
    if (ln == 0) Pm[(r + 8 * h) * TNUM + 16] = p1[r];
  }
  if (h == 0) Pm[16 * TNUM + ln] = p0[8];
  if (lane == 0) {
    Pm[16 * TNUM + 16] = p1[8];
    Pm[289] = ls;
  }
}

// ---------------------------------------------------------------------------
// Combine: one wave per batch. a <- P_c * a over the chunks, then
// denom = logscale + log(sum_j a_j * exp(end_j)).
// ---------------------------------------------------------------------------
__global__ __launch_bounds__(32) void crf_combine(
    const float* __restrict__ em, const float* __restrict__ start_tr,
    const float* __restrict__ end_tr, const float* __restrict__ Pws,
    float* __restrict__ den, int nchunk) {
  const int b = blockIdx.x;
  const int lane = (int)threadIdx.x;
  const int j = (lane < TNUM) ? lane : 0;

  float al = start_tr[j] + em[(size_t)b * LSEQ * TNUM + j];
  if (lane >= TNUM) al = -1.0e30f;
  float m0 = al;
#pragma unroll
  for (int off = 16; off > 0; off >>= 1) m0 = fmaxf(m0, __shfl_xor(m0, off, 32));
  float a = (lane < TNUM) ? __expf(al - m0) : 0.f;
  float ls = m0;

  for (int c = 0; c < nchunk; ++c) {
    const float* Pm = Pws + (size_t)(b * nchunk + c) * 290;
    const float* rowp = Pm + j * TNUM;  // clamped row -> always in bounds
    float s = 0.f;
#pragma unroll
    for (int k = 0; k < TNUM; ++k) {
      const float ak = __shfl(a, k, 32);
      s = fmaf(rowp[k], ak, s);
    }
    a = (lane < TNUM) ? s : 0.f;
    ls += Pm[289];
    float mx = a;
#pragma unroll
    for (int off = 16; off > 0; off >>= 1)
      mx = fmaxf(mx, __shfl_xor(mx, off, 32));
    ls += __logf(mx);
    a *= (1.f / mx);
  }

  float ee = __expf(end_tr[j]);
  if (lane >= TNUM) ee = 0.f;
  float v = a * ee;
#pragma unroll
  for (int off = 16; off > 0; off >>= 1) v += __shfl_xor(v, off, 32);
  if (lane == 0) den[b] = ls + __logf(v);
}

// ---------------------------------------------------------------------------
// Numerator: one block per batch, gather + reduce.
// ---------------------------------------------------------------------------
__global__ __launch_bounds__(256) void crf_num(
    const float* __restrict__ em, const int* __restrict__ tags,
    const unsigned char* __restrict__ mask, const float* __restrict__ start_tr,
    const float* __restrict__ end_tr, const float* __restrict__ trans,
    float* __restrict__ num) {
  __shared__ float s_tr[TNUM * TNUM];
  __shared__ float sacc[256];
  __shared__ int smsk[256];
  const int b = blockIdx.x, tid = (int)threadIdx.x;
  for (int i = tid; i < TNUM * TNUM; i += 256) s_tr[i] = trans[i];
  __syncthreads();

  const int* tg = tags + (size_t)b * LSEQ;
  const unsigned char* mk = mask + (size_t)b * LSEQ;
  const float* emb = em + (size_t)b * LSEQ * TNUM;

  float acc = 0.f;
  int msum = 0;
  for (int t = tid; t < LSEQ; t += 256) {
    const int tt = tg[t];
    const int m = mk[t] ? 1 : 0;
    msum += m;
    if (t == 0)
      acc += start_tr[tt] + emb[tt];
    else
      acc += (s_tr[tg[t - 1] * TNUM + tt] + emb[(size_t)t * TNUM + tt]) *
             (float)m;
  }
  sacc[tid] = acc;
  smsk[tid] = msum;
  __syncthreads();
  for (int s = 128; s > 0; s >>= 1) {
    if (tid < s) {
      sacc[tid] += sacc[tid + s];
      smsk[tid] += smsk[tid + s];
    }
    __syncthreads();
  }
  if (tid == 0) {
    const int se = smsk[0] - 1;  // seq_ends
    num[b] = sacc[0] + end_tr[tg[se]];
  }
}

// ---------------------------------------------------------------------------
// Final: out = -mean(num - den)
// ---------------------------------------------------------------------------
__global__ __launch_bounds__(256) void crf_final(const float* __restrict__ num,
                                                 const float* __restrict__ den,
                                                 float* __restrict__ out,
                                                 int nB) {
  __shared__ float s[256];
  const int tid = (int)threadIdx.x;
  float v = 0.f;
  for (int i = tid; i < nB; i += 256) v += num[i] - den[i];
  s[tid] = v;
  __syncthreads();
  for (int st = 128; st > 0; st >>= 1) {
    if (tid < st) s[tid] += s[tid + st];
    __syncthreads();
  }
  if (tid == 0) out[0] = -(s[0] / (float)nB);
}

extern "C" void kernel_launch(void* const* d_in, const int* in_sizes, int n_in,
                              void* d_out, int out_size, void* d_ws,
                              size_t ws_size, hipStream_t stream) {
  (void)n_in; (void)out_size;
  const float* em = (const float*)d_in[0];
  const int* tags = (const int*)d_in[1];            // JAX x64 off -> int32
  const unsigned char* mask = (const unsigned char*)d_in[2];
  const float* stp = (const float*)d_in[3];
  const float* enp = (const float*)d_in[4];
  const float* trp = (const float*)d_in[5];

  const int nB = in_sizes[1] / LSEQ;  // tags flat = B*L

  // Pick chunk count (parallelism) based on available workspace.
  int nchunk = 8;
  for (int cand = 32; cand >= 8; cand >>= 1) {
    size_t need = 8192 + (size_t)nB * cand * 290 * sizeof(float);
    if (need <= ws_size) { nchunk = cand; break; }
  }
  const int clen = (LSEQ - 1 + nchunk - 1) / nchunk;

  float* num = (float*)d_ws;
  float* den = num + nB;
  float* Pws = (float*)d_ws + 2048;  // 8 KB in, then nB*nchunk*290 floats

  crf_num<<<nB, 256, 0, stream>>>(em, tags, mask, stp, enp, trp, num);
  dim3 grid(nB, nchunk);
  crf_chunk<<<grid, 32, 0, stream>>>(em, mask, trp, Pws, clen);
  crf_combine<<<nB, 32, 0, stream>>>(em, stp, enp, Pws, den, nchunk);
  crf_final<<<1, 256, 0, stream>>>(num, den, (float*)d_out, nB);
}